// LearnableOLMEncoder_80350248173726
// MI455X (gfx1250) — compile-verified
//
#include <hip/hip_runtime.h>
#include <hip/hip_bf16.h>

#define NE (512 * 512)   // spatial elements
#define NL 256           // levels (softmax width)

#ifndef __has_builtin
#define __has_builtin(x) 0
#endif

#if defined(__gfx1250__) && \
    __has_builtin(__builtin_amdgcn_global_load_async_to_lds_b128) && \
    __has_builtin(__builtin_amdgcn_s_wait_asynccnt)
#define USE_ASYNC 1
#else
#define USE_ASYNC 0
#endif

#define AS1 __attribute__((address_space(1)))
#define AS3 __attribute__((address_space(3)))

typedef int   v4i __attribute__((vector_size(16)));      // matches builtin param type
typedef float v4f __attribute__((ext_vector_type(4)));   // native vector for NT store

// Stage one 1KB gumbel row (256 f32) into this wave's private LDS slice.
// 32 lanes x 16B per b128 op = 512B per instruction -> 2 instructions per row.
__device__ __forceinline__ void wave_row_to_lds(const float* grow, float* lbuf, int lane) {
#if USE_ASYNC
  char* g = (char*)grow + lane * 16;
  char* l = (char*)lbuf + lane * 16;
  __builtin_amdgcn_global_load_async_to_lds_b128((AS1 v4i*)g,         (AS3 v4i*)l,         0, 0);
  __builtin_amdgcn_global_load_async_to_lds_b128((AS1 v4i*)(g + 512), (AS3 v4i*)(l + 512), 0, 0);
#else
  (void)grow; (void)lbuf; (void)lane;
#endif
}

__global__ __launch_bounds__(256) void olm_gumbel_kernel(
    const int* __restrict__ qv,        // [NE] int32
    const float* __restrict__ logits,  // [256][256] f32
    const float* __restrict__ gumbel,  // [NE][256] f32
    float* __restrict__ out)           // [NE][256] f32
{
  const int lane = threadIdx.x & 31;
  const int wib  = threadIdx.x >> 5;                 // wave in block [0,8)
  const int wid  = blockIdx.x * 8 + wib;             // global wave id
  const int nw   = gridDim.x * 8;                    // total waves

#if USE_ASYNC
  __shared__ float lds[8 * 2 * NL];                  // 8 waves x 2 buffers x 1KB = 16KB
  float* buf[2] = { &lds[(wib * 2 + 0) * NL], &lds[(wib * 2 + 1) * NL] };
  int cur = 0;
#endif

  int e = wid;
#if USE_ASYNC
  if (e < NE) wave_row_to_lds(gumbel + (size_t)e * NL, buf[0], lane);
#endif

  for (; e < NE; e += nw) {
    const int enext = e + nw;
#if USE_ASYNC
    if (enext < NE) {
      wave_row_to_lds(gumbel + (size_t)enext * NL, buf[cur ^ 1], lane);
      __builtin_amdgcn_s_wait_asynccnt(2);   // async loads retire in order: current row done
    } else {
      __builtin_amdgcn_s_wait_asynccnt(0);
    }
    const v4f* g4 = (const v4f*)buf[cur];
#else
    const v4f* g4 = (const v4f*)(gumbel + (size_t)e * NL);
#endif

    // Warm L2 for a future gumbel row (lowers to global_prefetch_b8).
    const int epf = e + 2 * nw;
    if (epf < NE) __builtin_prefetch(gumbel + (size_t)epf * NL + lane * 8, 0, 0);

    // Row index from quantized value: clamp(-128,127)+128 (wave-uniform).
    int q = qv[e];
    q = q < -128 ? -128 : (q > 127 ? 127 : q);
    const v4f* t4 = (const v4f*)(logits + (size_t)(q + 128) * NL);

    // Lane l owns columns [8l, 8l+8).
    v4f t0 = t4[2 * lane], t1 = t4[2 * lane + 1];
    v4f g0 = g4[2 * lane], g1 = g4[2 * lane + 1];

    float tl[8] = { t0.x, t0.y, t0.z, t0.w, t1.x, t1.y, t1.z, t1.w };
    float x[8]  = { t0.x + g0.x, t0.y + g0.y, t0.z + g0.z, t0.w + g0.w,
                    t1.x + g1.x, t1.y + g1.y, t1.z + g1.z, t1.w + g1.w };

    // --- wave max of (logits + gumbel) for stable softmax ---
    float m = x[0];
    #pragma unroll
    for (int j = 1; j < 8; ++j) m = fmaxf(m, x[j]);
    #pragma unroll
    for (int off = 16; off > 0; off >>= 1)
      m = fmaxf(m, __shfl_xor(m, off, 32));

    // --- argmax of logits (first-index tie-break, matches jnp.argmax) ---
    float bv = tl[0]; int bc = 8 * lane;
    #pragma unroll
    for (int j = 1; j < 8; ++j)
      if (tl[j] > bv) { bv = tl[j]; bc = 8 * lane + j; }
    #pragma unroll
    for (int off = 16; off > 0; off >>= 1) {
      float ov = __shfl_xor(bv, off, 32);
      int   oc = __shfl_xor(bc, off, 32);
      if (ov > bv || (ov == bv && oc < bc)) { bv = ov; bc = oc; }
    }
    const float hard = (float)bc;

    // --- sum of exp across the wave ---
    float ex[8], s = 0.f;
    #pragma unroll
    for (int j = 0; j < 8; ++j) { ex[j] = __expf(x[j] - m); s += ex[j]; }
    #pragma unroll
    for (int off = 16; off > 0; off >>= 1) s += __shfl_xor(s, off, 32);
    const float r = 1.0f / s;

    // --- straight-through output: (hard + soft) - soft, kept unfused ---
    v4f o0, o1;
    { float sf;
      sf = ex[0] * r; o0.x = (hard + sf) - sf;
      sf = ex[1] * r; o0.y = (hard + sf) - sf;
      sf = ex[2] * r; o0.z = (hard + sf) - sf;
      sf = ex[3] * r; o0.w = (hard + sf) - sf;
      sf = ex[4] * r; o1.x = (hard + sf) - sf;
      sf = ex[5] * r; o1.y = (hard + sf) - sf;
      sf = ex[6] * r; o1.z = (hard + sf) - sf;
      sf = ex[7] * r; o1.w = (hard + sf) - sf;
    }
    v4f* orow = (v4f*)(out + (size_t)e * NL);
    __builtin_nontemporal_store(o0, &orow[2 * lane]);       // write-once: bypass cache
    __builtin_nontemporal_store(o1, &orow[2 * lane + 1]);

#if USE_ASYNC
    cur ^= 1;
#endif
  }
}

extern "C" void kernel_launch(void* const* d_in, const int* in_sizes, int n_in,
                              void* d_out, int out_size, void* d_ws, size_t ws_size,
                              hipStream_t stream) {
  const int*   qv     = (const int*)d_in[0];    // quantized_values [512*512] int32
  const float* logits = (const float*)d_in[1];  // encoding_logits  [256*256] f32
  const float* gumbel = (const float*)d_in[2];  // gumbel_noise     [512*512*256] f32
  float* out = (float*)d_out;                   // codes            [512*512*256] f32

  dim3 grid(2048), block(256);                  // 16384 waves, 16 elements each
  hipLaunchKernelGGL(olm_gumbel_kernel, grid, block, 0, stream,
                     qv, logits, gumbel, out);
}